// GCN_60430189855413
// MI455X (gfx1250) — compile-verified
//
#include <hip/hip_runtime.h>
#include <hip/hip_bf16.h>
#include <math.h>

typedef __attribute__((ext_vector_type(2))) float v2f;
typedef __attribute__((ext_vector_type(8))) float v8f;

// ---------------------------------------------------------------------------
// C[MxN] = H[MxK] * W[KxN], row-major fp32, via V_WMMA_F32_16X16X4_F32.
// One wave per 16x16 tile.  REQUIRES M % 16 == 0, N % 16 == 0, K % 4 == 0
// (guaranteed by construction: M = 100000, N padded to 48 for the last layer)
// so every load/store is unconditional -> no exec-mask juggling in the loop.
// A 16x4 frag (ISA 7.12.2): lanes 0-15 -> K={0,1}, lanes 16-31 -> K={2,3},
// M = lane%16.  B symmetric.  D: VGPR r -> M = r + 8*(lane>=16), N = lane%16.
// ---------------------------------------------------------------------------
template <int K, int N>
__global__ void gemm_wmma_f32(const float* __restrict__ H,
                              const float* __restrict__ W,
                              float* __restrict__ S) {
  const int lane  = threadIdx.x;        // 0..31, full wave
  const int half  = lane >> 4;
  const int lm    = lane & 15;
  const int baseM = blockIdx.y << 4;
  const int baseN = blockIdx.x << 4;
  const int m     = baseM + lm;
  const int n     = baseN + lm;

  // per-lane base pointers (all offsets below are compile-time constants)
  const float* __restrict__ Arow = H + (size_t)m * K + (half << 1);
  const float* __restrict__ Bcol = W + (size_t)(half << 1) * N + n;

  v8f acc = {};
  #pragma unroll
  for (int k = 0; k < K; k += 4) {
    v2f a = *(const v2f*)(Arow + k);    // contiguous pair -> global_load_b64
    v2f b;
    b.x = Bcol[k * N];
    b.y = Bcol[k * N + N];
    acc = __builtin_amdgcn_wmma_f32_16x16x4_f32(
        /*neg_a=*/false, a, /*neg_b=*/false, b,
        /*c_mod=*/(short)0, acc, /*reuse_a=*/false, /*reuse_b=*/false);
  }

  float* __restrict__ out = S + (size_t)(baseM + (half << 3)) * N + n;
  #pragma unroll
  for (int r = 0; r < 8; ++r) out[r * N] = acc[r];
}

// ---------------------------------------------------------------------------
// Sparse aggregation: agg[row[e], 0:NCH] += support[col[e], 0:NCH] (stride
// SSTR in the source).  EPB edges per block, blockDim = NCH*EPB.  Coalesced
// row gather + hardware fp32 atomics; support lives in L2 (51MB < 192MB).
// ---------------------------------------------------------------------------
template <int NCH, int SSTR, int EPB>
__global__ void scatter_add_edges(const float* __restrict__ S,
                                  const int* __restrict__ row,
                                  const int* __restrict__ col,
                                  float* __restrict__ agg, int E) {
  const int sub = threadIdx.x / NCH;    // compile-time NCH -> cheap
  const int t   = threadIdx.x % NCH;
  const int e   = blockIdx.x * EPB + sub;
  if (e >= E) return;
  const int r = row[e];
  const int c = col[e];
  unsafeAtomicAdd(&agg[(size_t)r * NCH + t], S[(size_t)c * SSTR + t]);
}

// ---------------------------------------------------------------------------
// Layer-1 post-aggregation: ReLU into h + per-channel sum / sumsq
// (block-local over 64 rows, then one atomic per channel per block).
// ---------------------------------------------------------------------------
__global__ void relu_stats128(const float* __restrict__ agg,
                              float* __restrict__ h,
                              float* __restrict__ sum,
                              float* __restrict__ sumsq, int n) {
  const int c  = threadIdx.x;           // channel, 0..127
  const int r0 = blockIdx.x * 64;
  float s = 0.0f, ss = 0.0f;
  for (int i = 0; i < 64; ++i) {
    const int r = r0 + i;
    if (r < n) {
      float v = agg[(size_t)r * 128 + c];
      v = v > 0.0f ? v : 0.0f;
      h[(size_t)r * 128 + c] = v;
      s  += v;
      ss += v * v;
    }
  }
  unsafeAtomicAdd(&sum[c],   s);
  unsafeAtomicAdd(&sumsq[c], ss);
}

// PairNorm stats: mu_c = sum_c/n;  mean((h-mu)^2) = mean_c(E[h^2]_c - mu_c^2).
__global__ void finalize_stats(float* __restrict__ sum,
                               float* __restrict__ sumsq,
                               float* __restrict__ inv, int n) {
  const int c = threadIdx.x;            // 0..127
  const float mu  = sum[c] / (float)n;
  const float ex2 = sumsq[c] / (float)n;
  sum[c] = mu;                          // keep mu for apply pass
  __shared__ float red[128];
  red[c] = ex2 - mu * mu;
  __syncthreads();
  for (int s = 64; s > 0; s >>= 1) {
    if (c < s) red[c] += red[c + s];
    __syncthreads();
  }
  if (c == 0) inv[0] = rsqrtf(red[0] / 128.0f);
}

// h = (h - mu[c]) * inv, float4-vectorized (channels are i*4 .. i*4+3).
__global__ void pairnorm_apply4(float4* __restrict__ h,
                                const float* __restrict__ mu,
                                const float* __restrict__ inv, int total4) {
  const int i = blockIdx.x * blockDim.x + threadIdx.x;
  if (i < total4) {
    const int c = (i << 2) & 127;
    const float iv = inv[0];
    float4 v = h[i];
    v.x = (v.x - mu[c + 0]) * iv;
    v.y = (v.y - mu[c + 1]) * iv;
    v.z = (v.z - mu[c + 2]) * iv;
    v.w = (v.w - mu[c + 3]) * iv;
    h[i] = v;
  }
}

__global__ void relu_copy4(const float4* __restrict__ src,
                           float4* __restrict__ dst, int total4) {
  const int i = blockIdx.x * blockDim.x + threadIdx.x;
  if (i < total4) {
    float4 v = src[i];
    v.x = v.x > 0.0f ? v.x : 0.0f;
    v.y = v.y > 0.0f ? v.y : 0.0f;
    v.z = v.z > 0.0f ? v.z : 0.0f;
    v.w = v.w > 0.0f ? v.w : 0.0f;
    dst[i] = v;
  }
}

// Zero-pad W_out (32x40) into Wpad (32x48) so the layer-4 GEMM has no guards.
__global__ void pad_w_48(const float* __restrict__ W, float* __restrict__ Wp) {
  const int i = blockIdx.x * blockDim.x + threadIdx.x;
  if (i < 32 * 48) {
    const int r = i / 48, c = i % 48;
    Wp[i] = (c < 40) ? W[r * 40 + c] : 0.0f;
  }
}

// Row-wise log_softmax over 40 channels.
__global__ void logsoftmax40(const float* __restrict__ agg,
                             float* __restrict__ out, int n) {
  const int r = blockIdx.x * blockDim.x + threadIdx.x;
  if (r >= n) return;
  const float* z = agg + (size_t)r * 40;
  float mx = z[0];
  for (int c = 1; c < 40; ++c) mx = fmaxf(mx, z[c]);
  float s = 0.0f;
  for (int c = 0; c < 40; ++c) s += __expf(z[c] - mx);
  const float lse = mx + __logf(s);
  float* o = out + (size_t)r * 40;
  for (int c = 0; c < 40; ++c) o[c] = z[c] - lse;
}

// ---------------------------------------------------------------------------
extern "C" void kernel_launch(void* const* d_in, const int* in_sizes, int n_in,
                              void* d_out, int out_size, void* d_ws, size_t ws_size,
                              hipStream_t stream) {
  const float* x    = (const float*)d_in[0];
  const int*   ei   = (const int*)d_in[1];
  const float* W_in = (const float*)d_in[2];
  const float* W1   = (const float*)d_in[3];
  const float* W2   = (const float*)d_in[4];
  const float* Wout = (const float*)d_in[5];

  const int n = in_sizes[0] / 128;   // 100000 nodes (multiple of 16)
  const int E = in_sizes[1] / 2;     // 1.6M edges
  const int* row = ei;               // edge_index[0]
  const int* col = ei + E;           // edge_index[1]

  float* ws = (float*)d_ws;
  const size_t buf = (size_t)n * 128;
  float* h     = ws;                 // node features (<=128 ch)
  float* sup   = ws + buf;           // GEMM output (stride = Nch, 48 for L4)
  float* agg   = ws + 2 * buf;       // scatter accumulator
  float* stats = ws + 3 * buf;       // sum[128] (->mu), sumsq[128], inv[1]
  float* Wpad  = ws + 3 * buf + 512; // padded W_out, 32x48

  const int tilesM = n / 16;         // 6250

  // ---- Layer 1: conv(x, W_in) [128 -> 128], ReLU, PairNorm ----
  gemm_wmma_f32<128, 128><<<dim3(8, tilesM), 32, 0, stream>>>(x, W_in, sup);
  hipMemsetAsync(agg, 0, (size_t)n * 128 * sizeof(float), stream);
  scatter_add_edges<128, 128, 2><<<(E + 1) / 2, 256, 0, stream>>>(sup, row, col, agg, E);
  hipMemsetAsync(stats, 0, 257 * sizeof(float), stream);
  relu_stats128<<<(n + 63) / 64, 128, 0, stream>>>(agg, h, stats, stats + 128, n);
  finalize_stats<<<1, 128, 0, stream>>>(stats, stats + 128, stats + 256, n);
  {
    const int t4 = n * 128 / 4;
    pairnorm_apply4<<<(t4 + 255) / 256, 256, 0, stream>>>((float4*)h, stats, stats + 256, t4);
  }

  // ---- Layer 2: conv(h, W1) [128 -> 64], ReLU ----
  gemm_wmma_f32<128, 64><<<dim3(4, tilesM), 32, 0, stream>>>(h, W1, sup);
  hipMemsetAsync(agg, 0, (size_t)n * 64 * sizeof(float), stream);
  scatter_add_edges<64, 64, 4><<<(E + 3) / 4, 256, 0, stream>>>(sup, row, col, agg, E);
  {
    const int t4 = n * 64 / 4;
    relu_copy4<<<(t4 + 255) / 256, 256, 0, stream>>>((const float4*)agg, (float4*)h, t4);
  }

  // ---- Layer 3: conv(h, W2) [64 -> 32], ReLU ----
  gemm_wmma_f32<64, 32><<<dim3(2, tilesM), 32, 0, stream>>>(h, W2, sup);
  hipMemsetAsync(agg, 0, (size_t)n * 32 * sizeof(float), stream);
  scatter_add_edges<32, 32, 8><<<(E + 7) / 8, 256, 0, stream>>>(sup, row, col, agg, E);
  {
    const int t4 = n * 32 / 4;
    relu_copy4<<<(t4 + 255) / 256, 256, 0, stream>>>((const float4*)agg, (float4*)h, t4);
  }

  // ---- Layer 4: conv(h, W_out) [32 -> 40], log_softmax ----
  pad_w_48<<<(32 * 48 + 255) / 256, 256, 0, stream>>>(Wout, Wpad);
  gemm_wmma_f32<32, 48><<<dim3(3, tilesM), 32, 0, stream>>>(h, Wpad, sup);
  hipMemsetAsync(agg, 0, (size_t)n * 40 * sizeof(float), stream);
  scatter_add_edges<40, 48, 8><<<(E + 7) / 8, 320, 0, stream>>>(sup, row, col, agg, E);
  logsoftmax40<<<(n + 255) / 256, 256, 0, stream>>>(agg, (float*)d_out, n);
}